// SparsePolyTeacher_75694503625156
// MI455X (gfx1250) — compile-verified
//
#include <hip/hip_runtime.h>
#include <hip/hip_bf16.h>

// SparsePoly teacher: out[b] = xS@a + xS·(triu(b,1)@xS) + sum monomial terms
// B=32768, D=1024, K=64, N3=N4=N5=512.
// Memory-bound (~128MB of x traffic @ 23.3TB/s ~ 5.5us). Quadratic form uses
// v_wmma_f32_16x16x4_f32 (full f32 precision, matches reference).

#define BDIM 1024
#define KDIM 64
#define NMON 512
#define ROWS_PER_BLOCK 128   // 8 waves * 16 rows
#define THREADS 256

typedef __attribute__((ext_vector_type(2))) float v2f;
typedef __attribute__((ext_vector_type(8))) float v8f;

__global__ __launch_bounds__(THREADS)
void sparse_poly_kernel(const float* __restrict__ x,
                        const int*   __restrict__ S,
                        const float* __restrict__ a,
                        const float* __restrict__ b,
                        const int*   __restrict__ idx3, const float* __restrict__ c3,
                        const int*   __restrict__ idx4, const float* __restrict__ c4,
                        const int*   __restrict__ idx5, const float* __restrict__ c5,
                        float* __restrict__ out)
{
    __shared__ float xs[ROWS_PER_BLOCK][KDIM];   // 32 KB gathered x tile
    __shared__ float buS[KDIM][KDIM];            // 16 KB triu(b,1)
    __shared__ float quadS[ROWS_PER_BLOCK];      // per-row quad results
    __shared__ int   Ssh[KDIM];
    __shared__ float ash[KDIM];

    const int tid  = threadIdx.x;
    const int wave = tid >> 5;
    const int lane = tid & 31;
    const int lo   = lane & 15;       // N / M index within fragment
    const int hi   = lane >> 4;       // half-wave select
    const int rowBlockBase = blockIdx.x * ROWS_PER_BLOCK;

    // ---- stage S, a, triu(b,1) ----
    if (tid < KDIM) {
        Ssh[tid] = S[tid];
        ash[tid] = a[tid];
    }
    #pragma unroll
    for (int e = tid; e < KDIM * KDIM; e += THREADS) {
        const int i = e >> 6, j = e & 63;
        buS[i][j] = (j > i) ? b[e] : 0.0f;
    }
    __syncthreads();  // need Ssh before the gather below

    // ---- gather xS tile: coalesced-ish over j (S is sorted ascending) ----
    for (int e = tid; e < ROWS_PER_BLOCK * KDIM; e += THREADS) {
        const int r = e >> 6, j = e & 63;
        xs[r][j] = x[(size_t)(rowBlockBase + r) * BDIM + Ssh[j]];
    }
    __syncthreads();

    // ---- quadratic form via WMMA f32 16x16x4 ----
    // wave w owns rows [16w, 16w+16). y = xs_tile @ buS, then row-dot with xs.
    const int waveBase = wave * 16;
    float part[8];
    #pragma unroll
    for (int p = 0; p < 8; ++p) part[p] = 0.0f;

    #pragma unroll
    for (int nt = 0; nt < 4; ++nt) {            // N tiles of 16
        v8f acc = {};
        #pragma unroll
        for (int kc = 0; kc < 16; ++kc) {       // K chunks of 4
            // A 16x4 f32: lane l -> M=lo, K = v + 2*hi  (v = vgpr index)
            v2f A, Bf;
            A.x = xs[waveBase + lo][4 * kc + 2 * hi + 0];
            A.y = xs[waveBase + lo][4 * kc + 2 * hi + 1];
            // B 4x16 f32: lane l -> N=lo, K = v + 2*hi
            Bf.x = buS[4 * kc + 2 * hi + 0][16 * nt + lo];
            Bf.y = buS[4 * kc + 2 * hi + 1][16 * nt + lo];
            acc = __builtin_amdgcn_wmma_f32_16x16x4_f32(
                false, A, false, Bf, (short)0, acc, false, false);
        }
        // D 16x16 f32: vgpr p, lane l -> M = p + 8*hi, N = lo (+16*nt)
        #pragma unroll
        for (int p = 0; p < 8; ++p)
            part[p] += acc[p] * xs[waveBase + p + 8 * hi][16 * nt + lo];
    }

    // reduce the row dot across the 16 lanes of each half-wave
    #pragma unroll
    for (int p = 0; p < 8; ++p) {
        float s = part[p];
        s += __shfl_xor(s, 8);
        s += __shfl_xor(s, 4);
        s += __shfl_xor(s, 2);
        s += __shfl_xor(s, 1);
        if (lo == p) quadS[waveBase + p + 8 * hi] = s;
    }
    __syncthreads();

    // ---- per-row linear + monomial terms (threads 0..127 own one row) ----
    if (tid < ROWS_PER_BLOCK) {
        const int lr = tid;
        const float* xr = xs[lr];
        float accum = quadS[lr];

        #pragma unroll 8
        for (int j = 0; j < KDIM; ++j)
            accum += xr[j] * ash[j];

        for (int t = 0; t < NMON; ++t) {
            const int i0 = idx3[3 * t + 0], i1 = idx3[3 * t + 1], i2 = idx3[3 * t + 2];
            accum += c3[t] * xr[i0] * xr[i1] * xr[i2];
        }
        for (int t = 0; t < NMON; ++t) {
            const int i0 = idx4[4 * t + 0], i1 = idx4[4 * t + 1];
            const int i2 = idx4[4 * t + 2], i3 = idx4[4 * t + 3];
            accum += c4[t] * xr[i0] * xr[i1] * xr[i2] * xr[i3];
        }
        for (int t = 0; t < NMON; ++t) {
            const int i0 = idx5[5 * t + 0], i1 = idx5[5 * t + 1], i2 = idx5[5 * t + 2];
            const int i3 = idx5[5 * t + 3], i4 = idx5[5 * t + 4];
            accum += c5[t] * xr[i0] * xr[i1] * xr[i2] * xr[i3] * xr[i4];
        }
        out[rowBlockBase + lr] = accum;
    }
}

extern "C" void kernel_launch(void* const* d_in, const int* in_sizes, int n_in,
                              void* d_out, int out_size, void* d_ws, size_t ws_size,
                              hipStream_t stream) {
    const float* x    = (const float*)d_in[0];
    const int*   S    = (const int*)  d_in[1];
    const float* a    = (const float*)d_in[2];
    const float* b    = (const float*)d_in[3];
    const int*   idx3 = (const int*)  d_in[4];
    const float* c3   = (const float*)d_in[5];
    const int*   idx4 = (const int*)  d_in[6];
    const float* c4   = (const float*)d_in[7];
    const int*   idx5 = (const int*)  d_in[8];
    const float* c5   = (const float*)d_in[9];
    float* out = (float*)d_out;

    const int nblocks = out_size / ROWS_PER_BLOCK;   // 32768/128 = 256
    sparse_poly_kernel<<<nblocks, THREADS, 0, stream>>>(
        x, S, a, b, idx3, c3, idx4, c4, idx5, c5, out);
}